// GAT_27221502722597
// MI455X (gfx1250) — compile-verified
//
#include <hip/hip_runtime.h>
#include <math.h>

// ---------------------------------------------------------------------------
// 2-layer GAT forward for MI455X (gfx1250, wave32).
//  - Layer-1 GEMM (100000x128 @ 128x32) via V_WMMA_F32_16X16X4_F32 (full f32).
//  - Segment softmax via float atomic-max + fused exp/accumulate atomics
//    (all intermediate buffers fit in the 192MB L2, so atomics stay on-chip).
// ---------------------------------------------------------------------------

#define NEG_SLOPE 0.2f

typedef __attribute__((ext_vector_type(2))) float v2f;
typedef __attribute__((ext_vector_type(8))) float v8f;

__device__ __forceinline__ float lrelu(float v) {
    return v > 0.0f ? v : NEG_SLOPE * v;
}

// float atomic max via monotone int/uint punning:
//   val >= 0 : signed int max ; val < 0 : unsigned int min
__device__ __forceinline__ void atomicMaxF(float* addr, float val) {
    if (val >= 0.0f) {
        atomicMax((int*)addr, __float_as_int(val));
    } else {
        atomicMin((unsigned int*)addr, (unsigned int)__float_as_int(val));
    }
}

// ---------------------------------------------------------------------------
// K0: initialize reduction buffers (must run every call; harness poisons ws).
// ---------------------------------------------------------------------------
__global__ __launch_bounds__(256) void k_init(float* __restrict__ acc1,
                                              float* __restrict__ m1,
                                              float* __restrict__ denom1,
                                              float* __restrict__ acc2,
                                              float* __restrict__ m2,
                                              float* __restrict__ denom2,
                                              int n) {
    int i = blockIdx.x * blockDim.x + threadIdx.x;
    const float NEG_INF = __int_as_float(0xFF800000);
    if (i < 32 * n) acc1[i] = 0.0f;
    if (i < 2 * n) {
        m1[i] = NEG_INF;
        denom1[i] = 0.0f;
        acc2[i] = 0.0f;
    }
    if (i < n) {
        m2[i] = NEG_INF;
        denom2[i] = 0.0f;
    }
}

// ---------------------------------------------------------------------------
// K1: h1 = x @ W1  using V_WMMA_F32_16X16X4_F32.
// One wave32 computes a 16-row x 32-col output tile (two 16x16 accumulators).
// A (16x4 f32) layout (ISA 7.12.2): lanes 0-15 -> M=lane, V0=K0,V1=K1;
//                                   lanes 16-31 -> M=lane-16, V0=K2,V1=K3.
// B (4x16 f32) mirrors with N instead of M.
// C (16x16 f32): lane<16 -> N=lane, VGPR j = M=j; lane>=16 -> M=j+8.
// ---------------------------------------------------------------------------
__global__ __launch_bounds__(256) void k_gemm1_wmma(const float* __restrict__ x,
                                                    const float* __restrict__ W1,
                                                    float* __restrict__ h1,
                                                    int n) {
    const int wave = (blockIdx.x * blockDim.x + threadIdx.x) >> 5;
    const int lane = threadIdx.x & 31;
    const int row0 = wave * 16;
    if (row0 >= n) return;                       // uniform per wave (EXEC all-1s)

    const int m  = lane & 15;                    // M (for A) / N (for B) in-tile
    const int kh = lane >> 4;                    // K half-select (0 -> K0/K1, 1 -> K2/K3)
    int row = row0 + m;
    if (row >= n) row = n - 1;                   // clamp loads, keep EXEC full

    const float* __restrict__ xrow = x + (size_t)row * 128;

    v8f c0 = {};                                 // cols 0..15
    v8f c1 = {};                                 // cols 16..31

    #pragma unroll
    for (int k0 = 0; k0 < 128; k0 += 4) {
        const int k = k0 + 2 * kh;
        v2f a = *(const v2f*)(xrow + k);         // A[M][k], A[M][k+1] (8B aligned)
        v2f b0, b1;
        b0.x = W1[(size_t)k * 32 + m];
        b0.y = W1[(size_t)(k + 1) * 32 + m];
        b1.x = W1[(size_t)k * 32 + 16 + m];
        b1.y = W1[(size_t)(k + 1) * 32 + 16 + m];
        c0 = __builtin_amdgcn_wmma_f32_16x16x4_f32(false, a, false, b0,
                                                   (short)0, c0, false, false);
        c1 = __builtin_amdgcn_wmma_f32_16x16x4_f32(false, a, false, b1,
                                                   (short)0, c1, false, false);
    }

    const int mb = kh * 8;                       // row block within tile
    #pragma unroll
    for (int j = 0; j < 8; ++j) {
        const int r = row0 + mb + j;
        if (r < n) {
            h1[(size_t)r * 32 + m]      = c0[j];
            h1[(size_t)r * 32 + 16 + m] = c1[j];
        }
    }
}

// ---------------------------------------------------------------------------
// K2: per-node attention logits for layer 1 (2 heads x 16 feats).
// ---------------------------------------------------------------------------
__global__ __launch_bounds__(256) void k_alpha1(const float* __restrict__ h1,
                                                const float* __restrict__ asw,
                                                const float* __restrict__ adw,
                                                float* __restrict__ asrc,
                                                float* __restrict__ adst,
                                                int n) {
    int i = blockIdx.x * blockDim.x + threadIdx.x;
    if (i >= n) return;
    const float* hr = h1 + (size_t)i * 32;
    #pragma unroll
    for (int h = 0; h < 2; ++h) {
        float s = 0.0f, d = 0.0f;
        #pragma unroll
        for (int f = 0; f < 16; ++f) {
            const float v = hr[h * 16 + f];
            s += v * asw[h * 16 + f];
            d += v * adw[h * 16 + f];
        }
        asrc[i * 2 + h] = s;
        adst[i * 2 + h] = d;
    }
}

// ---------------------------------------------------------------------------
// K3: edge pass A (layer 1) — segment max of leaky-relu logits per dst.
// One thread per (edge, head). Edges [E, E+n) are the implicit self-loops.
// ---------------------------------------------------------------------------
__global__ __launch_bounds__(256) void k_edge_max1(const int* __restrict__ src,
                                                   const int* __restrict__ dst,
                                                   const float* __restrict__ asrc,
                                                   const float* __restrict__ adst,
                                                   float* __restrict__ m1,
                                                   int E, int n) {
    int t = blockIdx.x * blockDim.x + threadIdx.x;
    const int tot = 2 * (E + n);
    if (t >= tot) return;
    const int h = t & 1;
    const int e = t >> 1;
    int s, d;
    if (e < E) { s = src[e]; d = dst[e]; } else { s = d = e - E; }
    const float logit = lrelu(asrc[(size_t)s * 2 + h] + adst[(size_t)d * 2 + h]);
    atomicMaxF(&m1[(size_t)d * 2 + h], logit);
}

// ---------------------------------------------------------------------------
// K4: edge pass B (layer 1) — ex = exp(e - m[dst]); accumulate unnormalized
// numerator (16 feats) and denominator per dst via float atomics.
// ---------------------------------------------------------------------------
__global__ __launch_bounds__(256) void k_edge_acc1(const int* __restrict__ src,
                                                   const int* __restrict__ dst,
                                                   const float* __restrict__ asrc,
                                                   const float* __restrict__ adst,
                                                   const float* __restrict__ m1,
                                                   const float* __restrict__ h1,
                                                   float* __restrict__ denom1,
                                                   float* __restrict__ acc1,
                                                   int E, int n) {
    int t = blockIdx.x * blockDim.x + threadIdx.x;
    const int tot = 2 * (E + n);
    if (t >= tot) return;
    const int h = t & 1;
    const int e = t >> 1;
    int s, d;
    if (e < E) { s = src[e]; d = dst[e]; } else { s = d = e - E; }
    const float logit = lrelu(asrc[(size_t)s * 2 + h] + adst[(size_t)d * 2 + h]);
    const float ex = __expf(logit - m1[(size_t)d * 2 + h]);
    atomicAdd(&denom1[(size_t)d * 2 + h], ex);
    const float* hs = h1 + (size_t)s * 32 + h * 16;
    float* ad = acc1 + (size_t)d * 32 + h * 16;
    #pragma unroll
    for (int f = 0; f < 16; ++f) {
        atomicAdd(&ad[f], ex * hs[f]);
    }
}

// ---------------------------------------------------------------------------
// K5: finalize layer 1 (normalize + bias + ReLU) and fused layer-2 projection
// (32x2 GEMV in registers) + layer-2 attention logits.
// ---------------------------------------------------------------------------
__global__ __launch_bounds__(256) void k_node_layer2(const float* __restrict__ acc1,
                                                     const float* __restrict__ denom1,
                                                     const float* __restrict__ b1,
                                                     const float* __restrict__ W2,
                                                     const float* __restrict__ as2w,
                                                     const float* __restrict__ ad2w,
                                                     float* __restrict__ g,
                                                     float* __restrict__ asrc2,
                                                     float* __restrict__ adst2,
                                                     int n) {
    int i = blockIdx.x * blockDim.x + threadIdx.x;
    if (i >= n) return;
    float g0 = 0.0f, g1 = 0.0f;
    const float inv0 = 1.0f / (denom1[(size_t)i * 2 + 0] + 1e-16f);
    const float inv1 = 1.0f / (denom1[(size_t)i * 2 + 1] + 1e-16f);
    #pragma unroll
    for (int c = 0; c < 32; ++c) {
        const float inv = (c < 16) ? inv0 : inv1;
        float v = acc1[(size_t)i * 32 + c] * inv + b1[c];
        v = v > 0.0f ? v : 0.0f;                 // ReLU between layers
        g0 += v * W2[c * 2 + 0];
        g1 += v * W2[c * 2 + 1];
    }
    g[(size_t)i * 2 + 0] = g0;
    g[(size_t)i * 2 + 1] = g1;
    asrc2[i] = g0 * as2w[0] + g1 * as2w[1];
    adst2[i] = g0 * ad2w[0] + g1 * ad2w[1];
}

// ---------------------------------------------------------------------------
// K6/K7: layer-2 edge passes (1 head, 2 features).
// ---------------------------------------------------------------------------
__global__ __launch_bounds__(256) void k_edge_max2(const int* __restrict__ src,
                                                   const int* __restrict__ dst,
                                                   const float* __restrict__ asrc2,
                                                   const float* __restrict__ adst2,
                                                   float* __restrict__ m2,
                                                   int E, int n) {
    int e = blockIdx.x * blockDim.x + threadIdx.x;
    if (e >= E + n) return;
    int s, d;
    if (e < E) { s = src[e]; d = dst[e]; } else { s = d = e - E; }
    atomicMaxF(&m2[d], lrelu(asrc2[s] + adst2[d]));
}

__global__ __launch_bounds__(256) void k_edge_acc2(const int* __restrict__ src,
                                                   const int* __restrict__ dst,
                                                   const float* __restrict__ asrc2,
                                                   const float* __restrict__ adst2,
                                                   const float* __restrict__ m2,
                                                   const float* __restrict__ g,
                                                   float* __restrict__ denom2,
                                                   float* __restrict__ acc2,
                                                   int E, int n) {
    int e = blockIdx.x * blockDim.x + threadIdx.x;
    if (e >= E + n) return;
    int s, d;
    if (e < E) { s = src[e]; d = dst[e]; } else { s = d = e - E; }
    const float ex = __expf(lrelu(asrc2[s] + adst2[d]) - m2[d]);
    atomicAdd(&denom2[d], ex);
    atomicAdd(&acc2[(size_t)d * 2 + 0], ex * g[(size_t)s * 2 + 0]);
    atomicAdd(&acc2[(size_t)d * 2 + 1], ex * g[(size_t)s * 2 + 1]);
}

// ---------------------------------------------------------------------------
// K8: final output.
// ---------------------------------------------------------------------------
__global__ __launch_bounds__(256) void k_out(const float* __restrict__ acc2,
                                             const float* __restrict__ denom2,
                                             const float* __restrict__ b2,
                                             float* __restrict__ out,
                                             int n) {
    int i = blockIdx.x * blockDim.x + threadIdx.x;
    if (i >= n) return;
    const float inv = 1.0f / (denom2[i] + 1e-16f);
    out[(size_t)i * 2 + 0] = acc2[(size_t)i * 2 + 0] * inv + b2[0];
    out[(size_t)i * 2 + 1] = acc2[(size_t)i * 2 + 1] * inv + b2[1];
}

// ---------------------------------------------------------------------------
extern "C" void kernel_launch(void* const* d_in, const int* in_sizes, int n_in,
                              void* d_out, int out_size, void* d_ws, size_t ws_size,
                              hipStream_t stream) {
    const float* x    = (const float*)d_in[0];
    const int*   ei   = (const int*)d_in[1];
    const float* W1   = (const float*)d_in[2];
    const float* as1w = (const float*)d_in[3];
    const float* ad1w = (const float*)d_in[4];
    const float* b1   = (const float*)d_in[5];
    const float* W2   = (const float*)d_in[6];
    const float* as2w = (const float*)d_in[7];
    const float* ad2w = (const float*)d_in[8];
    const float* b2   = (const float*)d_in[9];

    const int N = in_sizes[0] / 128;
    const int E = in_sizes[1] / 2;
    const int* srcI = ei;
    const int* dstI = ei + E;

    // workspace layout (floats)
    float* p = (float*)d_ws;
    float* h1     = p; p += (size_t)32 * N;
    float* asrc1  = p; p += (size_t)2 * N;
    float* adst1  = p; p += (size_t)2 * N;
    float* m1     = p; p += (size_t)2 * N;
    float* denom1 = p; p += (size_t)2 * N;
    float* acc1   = p; p += (size_t)32 * N;
    float* g      = p; p += (size_t)2 * N;
    float* asrc2  = p; p += (size_t)N;
    float* adst2  = p; p += (size_t)N;
    float* m2     = p; p += (size_t)N;
    float* denom2 = p; p += (size_t)N;
    float* acc2   = p; p += (size_t)2 * N;

    const int T = 256;
    const int Etot = E + N;

    // init reductions
    k_init<<<(32 * N + T - 1) / T, T, 0, stream>>>(acc1, m1, denom1, acc2, m2,
                                                   denom2, N);
    // layer-1 GEMM via WMMA: one wave per 16-row tile, 8 waves per block
    {
        const int tiles = (N + 15) / 16;
        const int waves_per_block = T / 32;
        const int blocks = (tiles + waves_per_block - 1) / waves_per_block;
        k_gemm1_wmma<<<blocks, T, 0, stream>>>(x, W1, h1, N);
    }
    k_alpha1<<<(N + T - 1) / T, T, 0, stream>>>(h1, as1w, ad1w, asrc1, adst1, N);
    k_edge_max1<<<(2 * Etot + T - 1) / T, T, 0, stream>>>(srcI, dstI, asrc1,
                                                          adst1, m1, E, N);
    k_edge_acc1<<<(2 * Etot + T - 1) / T, T, 0, stream>>>(srcI, dstI, asrc1,
                                                          adst1, m1, h1, denom1,
                                                          acc1, E, N);
    k_node_layer2<<<(N + T - 1) / T, T, 0, stream>>>(acc1, denom1, b1, W2, as2w,
                                                     ad2w, g, asrc2, adst2, N);
    k_edge_max2<<<(Etot + T - 1) / T, T, 0, stream>>>(srcI, dstI, asrc2, adst2,
                                                      m2, E, N);
    k_edge_acc2<<<(Etot + T - 1) / T, T, 0, stream>>>(srcI, dstI, asrc2, adst2,
                                                      m2, g, denom2, acc2, E, N);
    k_out<<<(N + T - 1) / T, T, 0, stream>>>(acc2, denom2, b2, (float*)d_out, N);
}